// GNN_35218731827641
// MI455X (gfx1250) — compile-verified
//
#include <hip/hip_runtime.h>
#include <hip/hip_bf16.h>

typedef __attribute__((ext_vector_type(2))) float v2f;
typedef __attribute__((ext_vector_type(8))) float v8f;

// ---------------------------------------------------------------------------
// Degree / normalization
// ---------------------------------------------------------------------------
__global__ void gcn_deg_init(float* __restrict__ deg, int n) {
    int i = blockIdx.x * blockDim.x + threadIdx.x;
    if (i < n) deg[i] = 1.0f;  // self loop contributes 1 to every node
}

__global__ void gcn_deg_acc(const int* __restrict__ dst, float* __restrict__ deg, int ne) {
    int e = blockIdx.x * blockDim.x + threadIdx.x;
    if (e < ne) atomicAdd(&deg[dst[e]], 1.0f);
}

__global__ void gcn_dinv(const float* __restrict__ deg, float* __restrict__ dinv, int n) {
    int i = blockIdx.x * blockDim.x + threadIdx.x;
    if (i < n) dinv[i] = rsqrtf(deg[i]);  // deg >= 1 always (self loops)
}

// ---------------------------------------------------------------------------
// Layer-1 skinny transform: T[n,128] = x[n,3] @ W1[3,128]
// ---------------------------------------------------------------------------
__global__ void gcn_gemm_in3(const float* __restrict__ x, const float* __restrict__ W1,
                             float* __restrict__ T, int n) {
    int tid = blockIdx.x * blockDim.x + threadIdx.x;
    if (tid >= n * 128) return;
    int i = tid >> 7;
    int f = tid & 127;
    const float* xr = x + i * 3;
    T[tid] = fmaf(xr[0], W1[f], fmaf(xr[1], W1[128 + f], xr[2] * W1[256 + f]));
}

// ---------------------------------------------------------------------------
// Dense transform T[n,128] = H[n,128] @ W[128,128] via V_WMMA_F32_16X16X4_F32.
// Block = 256 threads = 8 waves (wave32). Block handles 16 rows; wave w owns
// output columns [16w, 16w+16). K is walked in steps of 4 (32 WMMA ops).
// A 16x4 f32 lane layout: lanes 0-15 -> {K0,K1}, lanes 16-31 -> {K2,K3}.
// B 4x16 f32 lane layout: symmetric (N across lanes, K across halves/VGPRs).
// C/D 16x16 f32: VGPR v -> row v (lanes 0-15) / row v+8 (lanes 16-31).
// ---------------------------------------------------------------------------
__global__ __launch_bounds__(256) void gcn_gemm_wmma128(const float* __restrict__ H,
                                                        const float* __restrict__ W,
                                                        float* __restrict__ T, int n) {
    const int wave = threadIdx.x >> 5;   // 0..7 -> column tile
    const int lane = threadIdx.x & 31;
    const int half = lane >> 4;          // 0 or 1
    const int r    = lane & 15;
    const int m0   = blockIdx.x * 16;
    const int n0   = wave * 16;

    // Branchless row clamp keeps EXEC all-1s through the WMMA sequence.
    int row = m0 + r;
    row = row < n ? row : (n - 1);
    const float* ap = H + (long long)row * 128 + 2 * half;
    const float* bp = W + (long long)(2 * half) * 128 + n0 + r;

    v8f acc = {};
#pragma unroll
    for (int k0 = 0; k0 < 128; k0 += 4) {
        v2f a, b;
        a.x = ap[k0];
        a.y = ap[k0 + 1];
        b.x = bp[(long long)k0 * 128];
        b.y = bp[(long long)(k0 + 1) * 128];
        acc = __builtin_amdgcn_wmma_f32_16x16x4_f32(
            /*neg_a=*/false, a, /*neg_b=*/false, b,
            /*c_mod=*/(short)0, acc, /*reuse_a=*/false, /*reuse_b=*/false);
    }

#pragma unroll
    for (int v = 0; v < 8; ++v) {
        int orow = m0 + v + 8 * half;
        if (orow < n) T[(long long)orow * 128 + n0 + r] = acc[v];
    }
}

// ---------------------------------------------------------------------------
// Self-loop init: H[i,:] = T[i,:] * dinv[i]^2   (32 threads/node, float4 each)
// ---------------------------------------------------------------------------
__global__ void gcn_self_init(const float* __restrict__ T, const float* __restrict__ dinv,
                              float* __restrict__ H, int n) {
    long long tid = (long long)blockIdx.x * blockDim.x + threadIdx.x;
    int i  = (int)(tid >> 5);
    int f4 = ((int)tid & 31) * 4;
    if (i >= n) return;
    float w = dinv[i] * dinv[i];
    const float4 t = *(const float4*)(T + (long long)i * 128 + f4);
    float4 o;
    o.x = t.x * w; o.y = t.y * w; o.z = t.z * w; o.w = t.w * w;
    *(float4*)(H + (long long)i * 128 + f4) = o;
}

// ---------------------------------------------------------------------------
// Edge scatter: H[dst,:] += T[src,:] * dinv[src]*dinv[dst]
// 32 threads per edge, 4 features each -> GLOBAL_ATOMIC_ADD_F32 x4.
// ---------------------------------------------------------------------------
__global__ void gcn_scatter(const int* __restrict__ src, const int* __restrict__ dst,
                            const float* __restrict__ dinv, const float* __restrict__ T,
                            float* __restrict__ H, int ne) {
    long long tid = (long long)blockIdx.x * blockDim.x + threadIdx.x;
    long long e = tid >> 5;
    int f4 = ((int)tid & 31) * 4;
    if (e >= ne) return;
    int s = src[e];
    int d = dst[e];
    float w = dinv[s] * dinv[d];
    const float4 t = *(const float4*)(T + (long long)s * 128 + f4);
    float* out = H + (long long)d * 128 + f4;
    atomicAdd(out + 0, t.x * w);
    atomicAdd(out + 1, t.y * w);
    atomicAdd(out + 2, t.z * w);
    atomicAdd(out + 3, t.w * w);
}

// ---------------------------------------------------------------------------
// Bias + ReLU in place
// ---------------------------------------------------------------------------
__global__ void gcn_bias_relu(float* __restrict__ H, const float* __restrict__ b, int n) {
    long long tid = (long long)blockIdx.x * blockDim.x + threadIdx.x;
    if (tid >= (long long)n * 128) return;
    int f = (int)tid & 127;
    H[tid] = fmaxf(H[tid] + b[f], 0.0f);
}

// ---------------------------------------------------------------------------
// Global mean pool
// ---------------------------------------------------------------------------
__global__ void gcn_pool_zero(float* __restrict__ pooled, float* __restrict__ counts, int ng) {
    int i = blockIdx.x * blockDim.x + threadIdx.x;
    if (i < ng * 128) pooled[i] = 0.0f;
    if (i < ng) counts[i] = 0.0f;
}

__global__ void gcn_pool(const float* __restrict__ H, const int* __restrict__ batch,
                         float* __restrict__ pooled, float* __restrict__ counts, int n) {
    long long tid = (long long)blockIdx.x * blockDim.x + threadIdx.x;
    int i  = (int)(tid >> 5);
    int f4 = ((int)tid & 31) * 4;
    if (i >= n) return;
    int g = batch[i];
    const float4 h = *(const float4*)(H + (long long)i * 128 + f4);
    float* out = pooled + (long long)g * 128 + f4;
    atomicAdd(out + 0, h.x);
    atomicAdd(out + 1, h.y);
    atomicAdd(out + 2, h.z);
    atomicAdd(out + 3, h.w);
    if (f4 == 0) atomicAdd(&counts[g], 1.0f);
}

// ---------------------------------------------------------------------------
// Final FC: out[g,c] = (sum_f pooled[g,f] * Wfc[f,c]) / max(count,1) + bfc[c]
// ---------------------------------------------------------------------------
__global__ void gcn_fc(const float* __restrict__ pooled, const float* __restrict__ counts,
                       const float* __restrict__ Wfc, const float* __restrict__ bfc,
                       float* __restrict__ out, int ng) {
    int tid = blockIdx.x * blockDim.x + threadIdx.x;
    if (tid >= ng * 2) return;
    int g = tid >> 1;
    int c = tid & 1;
    float cnt = fmaxf(counts[g], 1.0f);
    const float* p = pooled + (long long)g * 128;
    float s = 0.0f;
#pragma unroll 4
    for (int f = 0; f < 128; ++f) s = fmaf(p[f], Wfc[f * 2 + c], s);
    out[g * 2 + c] = s / cnt + bfc[c];
}

// ---------------------------------------------------------------------------
// Host-side orchestration
// ---------------------------------------------------------------------------
static inline int div_up(long long a, long long b) { return (int)((a + b - 1) / b); }

extern "C" void kernel_launch(void* const* d_in, const int* in_sizes, int n_in,
                              void* d_out, int out_size, void* d_ws, size_t ws_size,
                              hipStream_t stream) {
    const float* x    = (const float*)d_in[0];
    const int*   eidx = (const int*)d_in[1];
    const int*   batch= (const int*)d_in[2];
    const float* W1   = (const float*)d_in[3];
    const float* b1   = (const float*)d_in[4];
    const float* W2   = (const float*)d_in[5];
    const float* b2   = (const float*)d_in[6];
    const float* W3   = (const float*)d_in[7];
    const float* b3   = (const float*)d_in[8];
    const float* Wfc  = (const float*)d_in[9];
    const float* bfc  = (const float*)d_in[10];
    float* out = (float*)d_out;

    const int n  = in_sizes[0] / 3;       // nodes
    const int ne = in_sizes[1] / 2;       // edges
    const int ng = out_size / 2;          // graphs
    const int* src = eidx;                // edge_index[0]
    const int* dst = eidx + ne;           // edge_index[1]

    // Workspace carve-out (all 256B aligned sections)
    char* ws = (char*)d_ws;
    auto carve = [&](size_t bytes) {
        void* p = (void*)ws;
        ws += (bytes + 255) & ~(size_t)255;
        return p;
    };
    float* T      = (float*)carve((size_t)n * 128 * sizeof(float));
    float* H      = (float*)carve((size_t)n * 128 * sizeof(float));
    float* deg    = (float*)carve((size_t)n * sizeof(float));
    float* dinv   = (float*)carve((size_t)n * sizeof(float));
    float* pooled = (float*)carve((size_t)ng * 128 * sizeof(float));
    float* counts = (float*)carve((size_t)ng * sizeof(float));

    const int B = 256;

    // Normalization
    gcn_deg_init<<<div_up(n, B), B, 0, stream>>>(deg, n);
    gcn_deg_acc <<<div_up(ne, B), B, 0, stream>>>(dst, deg, ne);
    gcn_dinv    <<<div_up(n, B), B, 0, stream>>>(deg, dinv, n);

    const long long nodeF4 = (long long)n * 32;   // 32 threads/node (4 feats each)
    const long long edgeF4 = (long long)ne * 32;

    // ---- Layer 1: T = x @ W1 (skinny), aggregate into H ----
    gcn_gemm_in3 <<<div_up((long long)n * 128, B), B, 0, stream>>>(x, W1, T, n);
    gcn_self_init<<<div_up(nodeF4, B), B, 0, stream>>>(T, dinv, H, n);
    gcn_scatter  <<<div_up(edgeF4, B), B, 0, stream>>>(src, dst, dinv, T, H, ne);
    gcn_bias_relu<<<div_up((long long)n * 128, B), B, 0, stream>>>(H, b1, n);

    // ---- Layer 2: T = H @ W2 (WMMA), aggregate into H ----
    gcn_gemm_wmma128<<<div_up(n, 16), B, 0, stream>>>(H, W2, T, n);
    gcn_self_init<<<div_up(nodeF4, B), B, 0, stream>>>(T, dinv, H, n);
    gcn_scatter  <<<div_up(edgeF4, B), B, 0, stream>>>(src, dst, dinv, T, H, ne);
    gcn_bias_relu<<<div_up((long long)n * 128, B), B, 0, stream>>>(H, b2, n);

    // ---- Layer 3: T = H @ W3 (WMMA), aggregate into H ----
    gcn_gemm_wmma128<<<div_up(n, 16), B, 0, stream>>>(H, W3, T, n);
    gcn_self_init<<<div_up(nodeF4, B), B, 0, stream>>>(T, dinv, H, n);
    gcn_scatter  <<<div_up(edgeF4, B), B, 0, stream>>>(src, dst, dinv, T, H, ne);
    gcn_bias_relu<<<div_up((long long)n * 128, B), B, 0, stream>>>(H, b3, n);

    // ---- Pool + FC ----
    gcn_pool_zero<<<div_up(ng * 128, B), B, 0, stream>>>(pooled, counts, ng);
    gcn_pool     <<<div_up(nodeF4, B), B, 0, stream>>>(H, batch, pooled, counts, n);
    gcn_fc       <<<div_up(ng * 2, B), B, 0, stream>>>(pooled, counts, Wfc, bfc, out, ng);
}